// CyclicNet_12154757448291
// MI455X (gfx1250) — compile-verified
//
#include <hip/hip_runtime.h>
#include <hip/hip_bf16.h>

typedef __attribute__((ext_vector_type(2))) float v2f;
typedef __attribute__((ext_vector_type(8))) float v8f;

#define BAG   15
#define NSTEP 5
#define LAMBDA 0.001f
#define PS    8
#define IM    28
#define DENOM (IM - PS)      // 20
#define BATCH 65536
#define INDIM 66             // PS*PS + 2
#define NBLK  512            // BATCH / 128
#define WISTR 68             // padded Wi row stride (K padded 66 -> 68)

// ---------------------------------------------------------------------------
// Prep kernel (1 block, 256 threads): build ReffT[k=j][n=i] = (R[i][j] -
// LAM * coact[i][j]) * mask[i][j] from the per-block coact partials of the
// previous step. Deterministic serial sum over the 512 block partials.
// ---------------------------------------------------------------------------
__global__ void cyclic_prep(const float* __restrict__ R,
                            const float* __restrict__ mask,
                            const float* __restrict__ partials,
                            float* __restrict__ ReffT,
                            int stepK) {
  int tid = threadIdx.x;            // 0..255
  int i = tid >> 4, j = tid & 15;   // coact indices
  float v = 0.f;
  if (stepK > 0 && i < BAG && j < BAG) {
    float co = 0.f;
    for (int b = 0; b < NBLK; ++b) co += partials[b * 256 + tid];
    co *= (1.0f / (float)BATCH);
    v = (R[i * BAG + j] - LAMBDA * co) * mask[i * BAG + j];
  }
  ReffT[j * 16 + i] = v;            // row = K (j), col = N (i)
}

// ---------------------------------------------------------------------------
// Main step kernel. 256 threads = 8 waves; each wave owns 16 batch rows and
// computes everything for them with V_WMMA_F32_16X16X4_F32 tiles. All small
// weights are staged zero-padded into LDS so fragment reads are branch-free.
// ---------------------------------------------------------------------------
__global__ __launch_bounds__(256)
void cyclic_step(const float* __restrict__ imgs,
                 const float* __restrict__ Wi,  const float* __restrict__ bi,
                 const float* __restrict__ gma, const float* __restrict__ bta,
                 const float* __restrict__ rb,
                 const float* __restrict__ Wc,  const float* __restrict__ bc,
                 const float* __restrict__ Wo,  const float* __restrict__ bo,
                 float* __restrict__ stateWS,   int* __restrict__ posWS,
                 float* __restrict__ partials,  const float* __restrict__ ReffT,
                 float* __restrict__ out,       int stepK) {
  __shared__ float ldsWi[16 * WISTR];     // Wi zero-padded to 16 x 68
  __shared__ float ldsWout[256];          // [k=feature][col] : Wo cols 0-9, Wc 10-11
  __shared__ float ldsReff[256];          // ReffT copy
  __shared__ float ldsPar[48];            // bias | gamma | beta, zero-padded to 16
  __shared__ float ldsState[8][16][16];   // per-wave new state tile [row][feat]
  __shared__ float ldsCo[8][256];         // per-wave coact partial tiles

  const int tid  = threadIdx.x;
  const int wave = tid >> 5;
  const int lane = tid & 31;
  const int half = lane >> 4;             // 0: K-lo / M 0..7 ; 1: K-hi / M 8..15
  const int n16  = lane & 15;             // N / feature index for C,D; row for A
  const int rowBase = blockIdx.x * 128 + wave * 16;
  const int myRow   = rowBase + n16;      // batch row this lane feeds A-frags for

  // ---- stage padded weights into LDS (branch-free fragment reads later)
  for (int idx = tid; idx < 16 * WISTR; idx += 256) {
    int n = idx / WISTR, k = idx - n * WISTR;
    ldsWi[idx] = (n < BAG && k < INDIM) ? Wi[n * INDIM + k] : 0.f;
  }
  {
    int k = tid >> 4, col = tid & 15;     // output-head weights
    float v = 0.f;
    if (k < BAG) {
      if (col < 10)      v = Wo[col * BAG + k];
      else if (col < 12) v = Wc[(col - 10) * BAG + k];
    }
    ldsWout[tid] = v;
    ldsReff[tid] = ReffT[tid];
  }
  if (tid < 16) {
    float bv = 0.f, gv = 0.f, btv = 0.f;
    if (tid < BAG) {
      bv  = bi[tid] + ((stepK > 0) ? rb[tid] : 0.f);
      gv  = gma[tid];
      btv = bta[tid];
    }
    ldsPar[tid]      = bv;
    ldsPar[16 + tid] = gv;
    ldsPar[32 + tid] = btv;
  }
  const float bc0 = bc[0], bc1 = bc[1];
  const float boN = bo[n16 < 10 ? n16 : 9];

  int p0, p1;
  if (stepK == 0) { p0 = DENOM / 2; p1 = DENOM / 2; }
  else            { p0 = posWS[2 * myRow]; p1 = posWS[2 * myRow + 1]; }

  const float* img = imgs + (size_t)myRow * (IM * IM);
  __syncthreads();                        // staged LDS ready

  // -------- in_act = [patch, pos] @ Wi.T : 17 chunks of K=4 (K padded to 68)
  v8f accI = {0.f, 0.f, 0.f, 0.f, 0.f, 0.f, 0.f, 0.f};
#pragma unroll
  for (int t = 0; t < 17; ++t) {
    const int kb = 4 * t + 2 * half;
    v2f a;
    if (t < 16) {                          // k = 0..63 : patch gather
      a.x = img[(p0 + (kb >> 3)) * IM + p1 + (kb & 7)];
      a.y = img[(p0 + ((kb + 1) >> 3)) * IM + p1 + ((kb + 1) & 7)];
    } else if (half == 0) {                // k = 64..65 : position features
      a.x = (float)p0 * (2.0f / DENOM) - 1.0f;
      a.y = (float)p1 * (2.0f / DENOM) - 1.0f;
    } else {                               // k = 66..67 : zero pad
      a.x = 0.f; a.y = 0.f;
    }
    v2f b = *(const v2f*)&ldsWi[n16 * WISTR + kb];   // zero-padded, aligned b64
    accI = __builtin_amdgcn_wmma_f32_16x16x4_f32(false, a, false, b,
                                                 (short)0, accI, false, false);
  }

  // -------- recurrent term: state @ Reff.T (K = 16, feature-padded)
  if (stepK > 0) {
    const float* st = stateWS + (size_t)myRow * 16;
#pragma unroll
    for (int t = 0; t < 4; ++t) {
      const int kb = 4 * t + 2 * half;
      v2f a, b;
      a.x = st[kb];
      a.y = st[kb + 1];
      b.x = ldsReff[kb * 16 + n16];
      b.y = ldsReff[(kb + 1) * 16 + n16];
      accI = __builtin_amdgcn_wmma_f32_16x16x4_f32(false, a, false, b,
                                                   (short)0, accI, false, false);
    }
  }

  // -------- bias + ReLU + LayerNorm over the 15 features of each row
  const float biasN = ldsPar[n16];        // 0 for pad column
  const float gN    = ldsPar[16 + n16];   // 0 for pad column -> LN output 0
  const float bN    = ldsPar[32 + n16];
#pragma unroll
  for (int r = 0; r < 8; ++r) {
    float h = fmaxf(accI[r] + biasN, 0.f);   // pad column: exactly 0
    float sum = h;
#pragma unroll
    for (int m = 8; m >= 1; m >>= 1) sum += __shfl_xor(sum, m, 16);
    float mu = sum * (1.0f / BAG);
    float dd = h - mu;
    float sq = (n16 < BAG) ? dd * dd : 0.f;  // exclude pad column from variance
#pragma unroll
    for (int m = 8; m >= 1; m >>= 1) sq += __shfl_xor(sq, m, 16);
    float s = dd * rsqrtf(sq * (1.0f / BAG) + 1e-5f) * gN + bN;  // pad -> 0
    const int M = r + half * 8;
    ldsState[wave][M][n16] = s;
    if (stepK < NSTEP - 1)
      stateWS[(size_t)(rowBase + M) * 16 + n16] = s;
  }
  __syncthreads();                        // ldsState tiles ready

  // -------- output head: cols 0..9 logits, 10..11 control (K = 16)
  v8f accO = {0.f, 0.f, 0.f, 0.f, 0.f, 0.f, 0.f, 0.f};
#pragma unroll
  for (int t = 0; t < 4; ++t) {
    const int kb = 4 * t + 2 * half;
    v2f a, b;
    a.x = ldsState[wave][n16][kb];        // A[m=row][k=feature]
    a.y = ldsState[wave][n16][kb + 1];
    b.x = ldsWout[kb * 16 + n16];
    b.y = ldsWout[(kb + 1) * 16 + n16];
    accO = __builtin_amdgcn_wmma_f32_16x16x4_f32(false, a, false, b,
                                                 (short)0, accO, false, false);
  }

  // control -> position update (not needed after the final step)
#pragma unroll
  for (int r = 0; r < 8; ++r) {
    float raw10 = __shfl(accO[r], 10, 16);
    float raw11 = __shfl(accO[r], 11, 16);
    int q0 = __shfl(p0, half * 8 + r, 16);   // pos of row (half*8+r)
    int q1 = __shfl(p1, half * 8 + r, 16);
    if (stepK < NSTEP - 1 && n16 == 0) {
      float c0 = tanhf(raw10 + bc0);
      float c1 = tanhf(raw11 + bc1);
      bool rowSel = fabsf(c0) >= fabsf(c1);
      int mv0 = rowSel ? ((c0 > 0.f) - (c0 < 0.f)) : 0;
      int mv1 = rowSel ? 0 : ((c1 > 0.f) - (c1 < 0.f));
      q0 = min(max(q0 + mv0, 0), DENOM);
      q1 = min(max(q1 + mv1, 0), DENOM);
      const int row = rowBase + half * 8 + r;
      posWS[2 * row]     = q0;
      posWS[2 * row + 1] = q1;
    }
  }

  // logits on the final step
  if (stepK == NSTEP - 1 && n16 < 10) {
#pragma unroll
    for (int r = 0; r < 8; ++r)
      out[(size_t)(rowBase + r + half * 8) * 10 + n16] = accO[r] + boN;
  }

  // -------- coact partial: state^T @ state — A-frag == B-frag (symmetric)
  if (stepK < NSTEP - 1) {
    v8f accC = {0.f, 0.f, 0.f, 0.f, 0.f, 0.f, 0.f, 0.f};
#pragma unroll
    for (int t = 0; t < 4; ++t) {
      const int kb = 4 * t + 2 * half;
      v2f f;
      f.x = ldsState[wave][kb][n16];      // S[k=batch][m/n=feature]
      f.y = ldsState[wave][kb + 1][n16];
      accC = __builtin_amdgcn_wmma_f32_16x16x4_f32(false, f, false, f,
                                                   (short)0, accC, false, false);
    }
#pragma unroll
    for (int r = 0; r < 8; ++r)
      ldsCo[wave][(r + half * 8) * 16 + n16] = accC[r];
  }
  __syncthreads();
  if (stepK < NSTEP - 1) {                // deterministic block reduction
    float acc = 0.f;
#pragma unroll
    for (int w = 0; w < 8; ++w) acc += ldsCo[w][tid];
    partials[(size_t)blockIdx.x * 256 + tid] = acc;
  }
}

// ---------------------------------------------------------------------------
extern "C" void kernel_launch(void* const* d_in, const int* in_sizes, int n_in,
                              void* d_out, int out_size, void* d_ws, size_t ws_size,
                              hipStream_t stream) {
  const float* x    = (const float*)d_in[0];
  const float* Wi   = (const float*)d_in[1];
  const float* bi   = (const float*)d_in[2];
  const float* gma  = (const float*)d_in[3];
  const float* bta  = (const float*)d_in[4];
  const float* R    = (const float*)d_in[5];
  const float* rb   = (const float*)d_in[6];
  const float* mask = (const float*)d_in[7];
  const float* Wc   = (const float*)d_in[8];
  const float* bc   = (const float*)d_in[9];
  const float* Wo   = (const float*)d_in[10];
  const float* bo   = (const float*)d_in[11];

  // workspace layout
  float* stateWS  = (float*)d_ws;                                   // B*16 f32
  int*   posWS    = (int*)((char*)d_ws + (size_t)BATCH * 16 * 4);   // B*2 i32
  float* partials = (float*)((char*)posWS + (size_t)BATCH * 2 * 4); // 512*256
  float* ReffT    = partials + (size_t)NBLK * 256;                  // 256

  for (int k = 0; k < NSTEP; ++k) {
    cyclic_prep<<<1, 256, 0, stream>>>(R, mask, partials, ReffT, k);
    cyclic_step<<<NBLK, 256, 0, stream>>>(x, Wi, bi, gma, bta, rb, Wc, bc,
                                          Wo, bo, stateWS, posWS, partials,
                                          ReffT, (float*)d_out, k);
  }
}